// SparseAttentionHighPass_65592740545202
// MI455X (gfx1250) — compile-verified
//
#include <hip/hip_runtime.h>
#include <hip/hip_bf16.h>

// N=50000 nodes, d=128, E=640000 edges.
#define NNODES 50000
#define NHID   128
#define NEDGES 640000
#define ALPHA  0.2f
#define INV_SQRT_D 0.08838834764831845f  // 1/sqrt(128)

typedef float v2f __attribute__((ext_vector_type(2)));
typedef float v8f __attribute__((ext_vector_type(8)));

// ---------------------------------------------------------------------------
// Workspace layout (floats):
//   Wb       : 3*8*32*64            = 49152    (B-fragment repacked weights)
//   Q        : NNODES*NHID          = 6400000
//   K        : NNODES*NHID          = 6400000
//   V        : NNODES*NHID          = 6400000
//   denom    : NNODES               = 50000
//   expsc    : NEDGES               = 640000
// total ~ 79.8 MB
// ---------------------------------------------------------------------------
#define WB_OFF    0
#define WB_ELEMS  (3 * 8 * 32 * 64)
#define Q_OFF     (WB_OFF + WB_ELEMS)
#define K_OFF     (Q_OFF + NNODES * NHID)
#define V_OFF     (K_OFF + NNODES * NHID)
#define DEN_OFF   (V_OFF + NNODES * NHID)
#define EXP_OFF   (DEN_OFF + NNODES)

// ---------------------------------------------------------------------------
// Zero-fill helper (output + denom must start at 0; harness poisons buffers).
// ---------------------------------------------------------------------------
__global__ void zero_f32_kernel(float* __restrict__ p, int n) {
    int i = blockIdx.x * blockDim.x + threadIdx.x;
    int stride = gridDim.x * blockDim.x;
    for (; i < n; i += stride) p[i] = 0.0f;
}

// ---------------------------------------------------------------------------
// Repack Wq/Wk/Wv into WMMA B-fragment order.
// B tile for V_WMMA_F32_16X16X4_F32 is 4x16 (KxN). Per ISA 32-bit layout:
//   lanes 0-15  : N = lane,    VGPR0 = K=0, VGPR1 = K=1
//   lanes 16-31 : N = lane-16, VGPR0 = K=2, VGPR1 = K=3
// Since Y = X @ W^T, B(k, n) = W[n][k].
// Wb[((mat*8 + nt)*32 + kk)*64 + lane*2 + j] = W_mat[nt*16 + n][4*kk + koff + j]
// ---------------------------------------------------------------------------
__global__ void repack_w_kernel(const float* __restrict__ Wq,
                                const float* __restrict__ Wk,
                                const float* __restrict__ Wv,
                                float* __restrict__ Wb) {
    int id = blockIdx.x * blockDim.x + threadIdx.x;   // one (mat,nt,kk,lane)
    if (id >= 3 * 8 * 32 * 32) return;
    int lane = id & 31;
    int kk   = (id >> 5) & 31;
    int nt   = (id >> 10) & 7;
    int mat  = id >> 13;
    int n    = lane & 15;
    int koff = (lane >> 4) << 1;
    const float* W = (mat == 0) ? Wq : (mat == 1) ? Wk : Wv;
    int k = 4 * kk + koff;
    const float* row = W + (nt * 16 + n) * NHID;
    Wb[id * 2 + 0] = row[k + 0];
    Wb[id * 2 + 1] = row[k + 1];
}

// ---------------------------------------------------------------------------
// Fused QKV GEMM: Q|K|V = X @ {Wq,Wk,Wv}^T using V_WMMA_F32_16X16X4_F32.
// Block: 256 threads (8 wave32). Each block computes 32 rows of all three
// 128-wide outputs = 2 mtiles x 3 mats x 8 ntiles = 48 16x16 tiles; each wave
// owns 6 tiles, accumulating over K=128 in 32 wmma steps.
// X tile staged in LDS via GLOBAL_LOAD_ASYNC_TO_LDS_B128 (ASYNCcnt path);
// rows padded 128->132 floats so A-frag reads are conflict-free ds_load_b64.
// ---------------------------------------------------------------------------
#define XPAD 132

__global__ void qkv_gemm_kernel(const float* __restrict__ X,
                                const float* __restrict__ Wb,
                                float* __restrict__ Q,
                                float* __restrict__ Kout,
                                float* __restrict__ V) {
    __shared__ float xs[32 * XPAD];

    const int tid  = threadIdx.x;
    const int row0 = blockIdx.x * 32;

    // Stage X[row0 .. row0+31][0..127] into LDS: 1024 chunks of 16B, 4 per
    // thread, moved by the async global->LDS engine (no VGPR round-trip).
    // 16B chunks never cross a row (128 floats = 32 chunks/row); LDS target
    // (r*132 + c)*4 bytes stays 16B-aligned since 132*4 = 528 = 33*16.
#pragma unroll
    for (int it = 0; it < 4; ++it) {
        int i = tid + 256 * it;          // chunk id 0 .. 1023
        int r = i >> 5;                  // row within tile
        int c = (i & 31) << 2;           // float column, multiple of 4
        if (row0 + r < NNODES) {
            unsigned ldsoff = (unsigned)(size_t)&xs[r * XPAD + c];
            const float* g = X + (size_t)(row0 + r) * NHID + c;
            asm volatile("global_load_async_to_lds_b128 %0, %1, off"
                         :: "v"(ldsoff), "v"(g)
                         : "memory");
        } else {
            float4 z = {0.0f, 0.0f, 0.0f, 0.0f};
            *(float4*)&xs[r * XPAD + c] = z;   // zero-pad OOB tail rows
        }
    }
    asm volatile("s_wait_asynccnt 0" ::: "memory");
    __syncthreads();

    const int wave = tid >> 5;
    const int lane = tid & 31;
    const int mrow = lane & 15;          // M index within A-frag
    const int koff = (lane >> 4) << 1;   // 0 or 2

    for (int tt = wave; tt < 48; tt += 8) {
        const int mat = tt >> 4;         // 0..2
        const int rem = tt & 15;
        const int mt  = rem >> 3;        // 0..1
        const int nt  = rem & 7;         // 0..7

        const float*  abase = &xs[(mt * 16 + mrow) * XPAD + koff];
        const float2* bbase = (const float2*)(Wb + ((mat * 8 + nt) * 32) * 64);

        v8f acc = {};
#pragma unroll
        for (int kk = 0; kk < 32; ++kk) {
            float2 af = *(const float2*)(abase + 4 * kk);   // ds_load_b64
            float2 bf = bbase[kk * 32 + lane];              // coalesced 8B/lane
            v2f a; a.x = af.x; a.y = af.y;
            v2f b; b.x = bf.x; b.y = bf.y;
            acc = __builtin_amdgcn_wmma_f32_16x16x4_f32(
                false, a, false, b, (short)0, acc, false, false);
        }

        float* out = (mat == 0) ? Q : (mat == 1) ? Kout : V;
        const int cn = nt * 16 + (lane & 15);
        const int rb = row0 + mt * 16 + ((lane >> 4) << 3);
#pragma unroll
        for (int j = 0; j < 8; ++j) {
            int r = rb + j;
            if (r < NNODES) out[r * NHID + cn] = acc[j];
        }
    }
}

// ---------------------------------------------------------------------------
// Edge scores: one wave per edge. Lanes hold float4 slices of the 128-dim
// rows (32 lanes * 4 = 128), tree-reduce with shfl_xor, lane 0 does the
// exp + scalar fp32 atomic into denom.
// ---------------------------------------------------------------------------
__global__ void edge_score_kernel(const float* __restrict__ Q,
                                  const float* __restrict__ K,
                                  const int* __restrict__ src,
                                  const int* __restrict__ dst,
                                  float* __restrict__ expsc,
                                  float* __restrict__ denom) {
    const int e = blockIdx.x * (blockDim.x >> 5) + (threadIdx.x >> 5);
    if (e >= NEDGES) return;
    const int lane = threadIdx.x & 31;
    const int s = src[e];
    const int d = dst[e];

    float4 q = ((const float4*)(Q + s * NHID))[lane];
    float4 k = ((const float4*)(K + d * NHID))[lane];
    float p = q.x * k.x + q.y * k.y + q.z * k.z + q.w * k.w;

#pragma unroll
    for (int off = 16; off > 0; off >>= 1)
        p += __shfl_xor(p, off, 32);

    if (lane == 0) {
        p *= INV_SQRT_D;
        p = (p > 0.0f) ? p : ALPHA * p;     // LeakyReLU(0.2)
        float ex = __expf(p);
        expsc[e] = ex;
        (void)__hip_atomic_fetch_add(&denom[s], ex, __ATOMIC_RELAXED,
                                     __HIP_MEMORY_SCOPE_AGENT);
    }
}

// ---------------------------------------------------------------------------
// Aggregate: one wave per edge; out[src] += attn * V[dst] with fp32 atomics.
// ---------------------------------------------------------------------------
__global__ void edge_aggregate_kernel(const float* __restrict__ V,
                                      const int* __restrict__ src,
                                      const int* __restrict__ dst,
                                      const float* __restrict__ expsc,
                                      const float* __restrict__ denom,
                                      float* __restrict__ out) {
    const int e = blockIdx.x * (blockDim.x >> 5) + (threadIdx.x >> 5);
    if (e >= NEDGES) return;
    const int lane = threadIdx.x & 31;
    const int s = src[e];
    const int d = dst[e];

    const float attn = expsc[e] / (denom[s] + 1e-16f);
    float4 v = ((const float4*)(V + d * NHID))[lane];
    float* o = out + s * NHID + lane * 4;
    (void)__hip_atomic_fetch_add(o + 0, attn * v.x, __ATOMIC_RELAXED, __HIP_MEMORY_SCOPE_AGENT);
    (void)__hip_atomic_fetch_add(o + 1, attn * v.y, __ATOMIC_RELAXED, __HIP_MEMORY_SCOPE_AGENT);
    (void)__hip_atomic_fetch_add(o + 2, attn * v.z, __ATOMIC_RELAXED, __HIP_MEMORY_SCOPE_AGENT);
    (void)__hip_atomic_fetch_add(o + 3, attn * v.w, __ATOMIC_RELAXED, __HIP_MEMORY_SCOPE_AGENT);
}

// ---------------------------------------------------------------------------
extern "C" void kernel_launch(void* const* d_in, const int* in_sizes, int n_in,
                              void* d_out, int out_size, void* d_ws, size_t ws_size,
                              hipStream_t stream) {
    const float* X   = (const float*)d_in[0];   // residual [N,128]
    const float* Wq  = (const float*)d_in[1];   // [128,128]
    const float* Wk  = (const float*)d_in[2];
    const float* Wv  = (const float*)d_in[3];
    const int*   ei  = (const int*)d_in[4];     // [2,E] flat
    const int*   src = ei;
    const int*   dst = ei + NEDGES;

    float* ws    = (float*)d_ws;
    float* Wb    = ws + WB_OFF;
    float* Q     = ws + Q_OFF;
    float* K     = ws + K_OFF;
    float* V     = ws + V_OFF;
    float* denom = ws + DEN_OFF;
    float* expsc = ws + EXP_OFF;
    float* out   = (float*)d_out;

    // 1) zero output + denom
    zero_f32_kernel<<<2048, 256, 0, stream>>>(out, NNODES * NHID);
    zero_f32_kernel<<<64, 256, 0, stream>>>(denom, NNODES);

    // 2) repack weights into WMMA B-fragment layout
    repack_w_kernel<<<(3 * 8 * 32 * 32 + 255) / 256, 256, 0, stream>>>(Wq, Wk, Wv, Wb);

    // 3) fused QKV GEMM (fp32 WMMA, async LDS staging)
    qkv_gemm_kernel<<<(NNODES + 31) / 32, 256, 0, stream>>>(X, Wb, Q, K, V);

    // 4) edge scores + softmax denominator
    edge_score_kernel<<<(NEDGES + 7) / 8, 256, 0, stream>>>(Q, K, src, dst, expsc, denom);

    // 5) attention-weighted aggregation
    edge_aggregate_kernel<<<(NEDGES + 7) / 8, 256, 0, stream>>>(V, src, dst, expsc, denom, out);
}